// PointNet2SetAbstraction_53506702574031
// MI455X (gfx1250) — compile-verified
//
#include <hip/hip_runtime.h>
#include <hip/hip_bf16.h>

// ---------------------------------------------------------------------------
// PointNet++ Set Abstraction for MI455X (gfx1250, wave32).
//   1) fps_kernel:   1 workgroup (1024 thr) per batch; points cached in VGPRs.
//   2) ballq_kernel: 1 wave32 per centroid; ballot+prefix-popcount ordered
//                    first-64 in-radius selection with early exit.
//   3) mlp_kernel:   gather + 3-layer pointwise MLP via V_WMMA_F32_16X16X4_F32
//                    with all fragments fetched as single aligned ds_load_b64
//                    (K-interleaved weight layout), + max-pool over samples.
// ---------------------------------------------------------------------------

namespace {
constexpr int kB = 8;
constexpr int kN = 16384;
constexpr int kNPoint = 512;
constexpr int kNSample = 64;
constexpr int kCin = 64;
constexpr float kR2 = 0.01f;   // RADIUS^2
}

typedef float v2f __attribute__((ext_vector_type(2)));
typedef float v8f __attribute__((ext_vector_type(8)));

// argmax with (value desc, index asc) tie-break -> matches jnp.argmax "first max"
__device__ __forceinline__ void argmax_red32(float& v, int& i) {
#pragma unroll
  for (int off = 16; off > 0; off >>= 1) {
    float ov = __shfl_down(v, off, 32);
    int   oi = __shfl_down(i, off, 32);
    if (ov > v || (ov == v && oi < i)) { v = ov; i = oi; }
  }
}

// ------------------------------ 1) FPS -------------------------------------
__global__ __launch_bounds__(1024) void fps_kernel(const float* __restrict__ xyz,
                                                   float* __restrict__ newxyz) {
  const int b = blockIdx.x;
  const int t = threadIdx.x;
  const float* P = xyz + (size_t)b * kN * 3;

  // 16 points per thread, fully register resident.
  float px[16], py[16], pz[16], dst[16];
#pragma unroll
  for (int k = 0; k < 16; ++k) {
    const int i = t + (k << 10);
    px[k] = P[i * 3 + 0];
    py[k] = P[i * 3 + 1];
    pz[k] = P[i * 3 + 2];
    dst[k] = 1e10f;
  }

  __shared__ float redV[32];
  __shared__ int   redI[32];
  __shared__ float sLX, sLY, sLZ;
  __shared__ int   sLI;

  if (t == 0) {
    sLX = P[0]; sLY = P[1]; sLZ = P[2];
    float* o = newxyz + (size_t)b * kNPoint * 3;
    o[0] = P[0]; o[1] = P[1]; o[2] = P[2];   // fps_idx[0] == 0
  }
  __syncthreads();

  const int wid = t >> 5, lane = t & 31;

  for (int j = 1; j < kNPoint; ++j) {
    const float lx = sLX, ly = sLY, lz = sLZ;
    float bv = -1.0f;
    int   bi = 0x7fffffff;
#pragma unroll
    for (int k = 0; k < 16; ++k) {
      const float dx = px[k] - lx, dy = py[k] - ly, dz = pz[k] - lz;
      const float d = dx * dx + dy * dy + dz * dz;
      const float nd = fminf(dst[k], d);
      dst[k] = nd;
      if (nd > bv) { bv = nd; bi = t + (k << 10); }  // k ascending => lowest idx on tie
    }
    argmax_red32(bv, bi);
    if (lane == 0) { redV[wid] = bv; redI[wid] = bi; }
    __syncthreads();
    if (wid == 0) {
      float v = redV[lane];
      int   i = redI[lane];
      argmax_red32(v, i);
      if (lane == 0) sLI = i;
    }
    __syncthreads();
    const int li = sLI;
    if (t == (li & 1023)) {                 // owner thread broadcasts coords
      const int k = li >> 10;
      sLX = px[k]; sLY = py[k]; sLZ = pz[k];
      float* o = newxyz + ((size_t)b * kNPoint + j) * 3;
      o[0] = px[k]; o[1] = py[k]; o[2] = pz[k];
    }
    __syncthreads();
  }
}

// --------------------------- 2) Ball query ---------------------------------
__global__ void ballq_kernel(const float* __restrict__ xyz,
                             const float* __restrict__ newxyz,
                             int* __restrict__ idxout) {
  const int gid = blockIdx.x * (blockDim.x >> 5) + (threadIdx.x >> 5);
  const int lane = threadIdx.x & 31;
  if (gid >= kB * kNPoint) return;
  const int b = gid >> 9;                                   // NPOINT = 512
  const float cx = newxyz[gid * 3 + 0];
  const float cy = newxyz[gid * 3 + 1];
  const float cz = newxyz[gid * 3 + 2];
  const float* P = xyz + (size_t)b * kN * 3;
  int* og = idxout + (size_t)gid * kNSample;

  int count = 0;
  int firstIdx = -1;
  for (int base = 0; base < kN; base += 32) {
    const int i = base + lane;
    const float dx = P[i * 3 + 0] - cx;
    const float dy = P[i * 3 + 1] - cy;
    const float dz = P[i * 3 + 2] - cz;
    const bool pred = (dx * dx + dy * dy + dz * dz) < kR2;
    const unsigned mask = __builtin_amdgcn_ballot_w32(pred);
    if (firstIdx < 0 && mask) firstIdx = base + __builtin_ctz(mask);
    if (pred) {
      const int pos = count + __popc(mask & ((1u << lane) - 1u));
      if (pos < kNSample) og[pos] = i;        // ordered insertion, ascending idx
    }
    count += __popc(mask);
    if (count >= kNSample) break;             // uniform across the wave
  }
  const int filled = count < kNSample ? count : kNSample;
  if (firstIdx < 0) firstIdx = 0;             // no hit -> pad with 0 (ref semantics)
  for (int p2 = filled + lane; p2 < kNSample; p2 += 32) og[p2] = firstIdx;
}

// --------------------- 3) Gather + MLP (WMMA) + maxpool --------------------
// Block = 128 threads (4 waves). Wave w owns rows [16w, 16w+16).
// A operands (Xs/H1/H2): row stride 68 floats -> conflict-free, 8B-aligned
//   contiguous K pairs -> single ds_load_b64 per fragment.
// B operands (weights): K-interleaved packing Wp[(k>>2)*4*ncols + c*4 + (k&3)]
//   -> lane's two K values adjacent -> single ds_load_b64, conflict-free.
__global__ __launch_bounds__(128) void mlp_kernel(
    const float* __restrict__ xyz, const float* __restrict__ features,
    const float* __restrict__ newxyz, const int* __restrict__ idxg,
    const float* __restrict__ W1, const float* __restrict__ b1,
    const float* __restrict__ W2, const float* __restrict__ b2,
    const float* __restrict__ W3, const float* __restrict__ b3,
    float* __restrict__ out) {
  __shared__ float Wp1[17 * 256];   // K=0..67 (row 67 zero), 64 cols, packed
  __shared__ float Wp2[16 * 256];   // K=0..63, 64 cols, packed
  __shared__ float Wp3[16 * 512];   // K=0..63, 128 cols, packed
  __shared__ float b1s[64], b2s[64], b3s[128];
  __shared__ float Xs[64 * 68];     // 64 samples x (3 xyz + 64 feat + pad)
  __shared__ float H1[64 * 68];
  __shared__ float H2[64 * 68];
  __shared__ float Pmax[4 * 128];

  const int t = threadIdx.x;

  // Stage weights/biases once per block (K-interleaved packing).
  for (int e = t; e < 68 * 64; e += 128) {
    const int k = e >> 6, c = e & 63;
    Wp1[(k >> 2) * 256 + c * 4 + (k & 3)] = (k < 67) ? W1[k * 64 + c] : 0.0f;
  }
  for (int e = t; e < 64 * 64; e += 128) {
    const int k = e >> 6, c = e & 63;
    Wp2[(k >> 2) * 256 + c * 4 + (k & 3)] = W2[e];
  }
  for (int e = t; e < 64 * 128; e += 128) {
    const int k = e >> 7, c = e & 127;
    Wp3[(k >> 2) * 512 + c * 4 + (k & 3)] = W3[e];
  }
  if (t < 64) { b1s[t] = b1[t]; b2s[t] = b2[t]; }
  if (t < 128) b3s[t] = b3[t];
  __syncthreads();

  const int w = t >> 5, lane = t & 31;
  const int m = lane & 15;       // N (or M for A-frag) within tile
  const int hi = lane >> 4;      // lane half
  const int ko = hi << 1;        // A/B fragment K offset: 0 or 2
  const int rb = w << 4;         // row-tile base for this wave
  const int ro = hi << 3;        // C-fragment row offset: 0 or 8

  for (int g = blockIdx.x; g < kB * kNPoint; g += (int)gridDim.x) {
    const int b = g >> 9;

    // ---- gather: X[s][0:3] = centered xyz, X[s][3:67] = features, X[s][67]=0
    {
      const int s = t & 63, half = t >> 6;
      const int n = idxg[(size_t)g * kNSample + s];
      const float c0 = newxyz[g * 3 + 0];
      const float c1 = newxyz[g * 3 + 1];
      const float c2 = newxyz[g * 3 + 2];
      const float* pc = xyz + ((size_t)b * kN + n) * 3;
      const float* pf = features + (size_t)b * kCin * kN + n;
      // thread covers element pairs (cp, cp+1), cp ≡ 2*half (mod 4) -> b64 stores
      for (int cp = half << 1; cp < 68; cp += 4) {
        v2f v;
#pragma unroll
        for (int u = 0; u < 2; ++u) {
          const int c = cp + u;
          float x;
          if (c == 0)      x = pc[0] - c0;
          else if (c == 1) x = pc[1] - c1;
          else if (c == 2) x = pc[2] - c2;
          else if (c < 67) x = pf[(size_t)(c - 3) * kN];
          else             x = 0.0f;
          v[u] = x;
        }
        *(v2f*)(Xs + s * 68 + cp) = v;
      }
    }
    __syncthreads();

    // ---- layer 1: Xs(64x68) @ W1(68x64) + b1, ReLU -> H1
#pragma unroll
    for (int j = 0; j < 4; ++j) {
      v8f acc;
      const float bvv = b1s[j * 16 + m];
#pragma unroll
      for (int r = 0; r < 8; ++r) acc[r] = bvv;
      const float* xr = Xs + (rb + m) * 68 + ko;
      const float* wr = Wp1 + (j * 16 + m) * 4 + ko;
#pragma unroll
      for (int kk = 0; kk < 17; ++kk) {
        const v2f a  = *(const v2f*)(xr + 4 * kk);
        const v2f bb = *(const v2f*)(wr + 256 * kk);
        acc = __builtin_amdgcn_wmma_f32_16x16x4_f32(false, a, false, bb,
                                                    (short)0, acc, false, false);
      }
#pragma unroll
      for (int r = 0; r < 8; ++r) {
        const float v = acc[r];
        H1[(rb + ro + r) * 68 + j * 16 + m] = v > 0.0f ? v : 0.0f;
      }
    }

    // ---- layer 2: H1(64x64) @ W2(64x64) + b2, ReLU -> H2  (wave-local rows)
#pragma unroll
    for (int j = 0; j < 4; ++j) {
      v8f acc;
      const float bvv = b2s[j * 16 + m];
#pragma unroll
      for (int r = 0; r < 8; ++r) acc[r] = bvv;
      const float* xr = H1 + (rb + m) * 68 + ko;
      const float* wr = Wp2 + (j * 16 + m) * 4 + ko;
#pragma unroll
      for (int kk = 0; kk < 16; ++kk) {
        const v2f a  = *(const v2f*)(xr + 4 * kk);
        const v2f bb = *(const v2f*)(wr + 256 * kk);
        acc = __builtin_amdgcn_wmma_f32_16x16x4_f32(false, a, false, bb,
                                                    (short)0, acc, false, false);
      }
#pragma unroll
      for (int r = 0; r < 8; ++r) {
        const float v = acc[r];
        H2[(rb + ro + r) * 68 + j * 16 + m] = v > 0.0f ? v : 0.0f;
      }
    }

    // ---- layer 3: H2(64x64) @ W3(64x128) + b3, ReLU, row-max per wave
#pragma unroll
    for (int j = 0; j < 8; ++j) {
      v8f acc;
      const float bvv = b3s[j * 16 + m];
#pragma unroll
      for (int r = 0; r < 8; ++r) acc[r] = bvv;
      const float* xr = H2 + (rb + m) * 68 + ko;
      const float* wr = Wp3 + (j * 16 + m) * 4 + ko;
#pragma unroll
      for (int kk = 0; kk < 16; ++kk) {
        const v2f a  = *(const v2f*)(xr + 4 * kk);
        const v2f bb = *(const v2f*)(wr + 512 * kk);
        acc = __builtin_amdgcn_wmma_f32_16x16x4_f32(false, a, false, bb,
                                                    (short)0, acc, false, false);
      }
      float mx = 0.0f;                       // ReLU outputs are >= 0
#pragma unroll
      for (int r = 0; r < 8; ++r) {
        const float v = acc[r];
        mx = fmaxf(mx, v > 0.0f ? v : 0.0f);
      }
      mx = fmaxf(mx, __shfl_xor(mx, 16, 32));  // combine rows r and r+8 halves
      if (hi == 0) Pmax[w * 128 + j * 16 + m] = mx;
    }
    __syncthreads();

    if (t < 128) {
      const float v = fmaxf(fmaxf(Pmax[t], Pmax[128 + t]),
                            fmaxf(Pmax[256 + t], Pmax[384 + t]));
      out[(size_t)g * 128 + t] = v;
    }
    __syncthreads();   // protect Xs / Pmax reuse by next group
  }
}

// ------------------------------ launcher -----------------------------------
extern "C" void kernel_launch(void* const* d_in, const int* in_sizes, int n_in,
                              void* d_out, int out_size, void* d_ws, size_t ws_size,
                              hipStream_t stream) {
  const float* xyz      = (const float*)d_in[0];
  const float* features = (const float*)d_in[1];
  const float* W1 = (const float*)d_in[2];
  const float* b1 = (const float*)d_in[3];
  const float* W2 = (const float*)d_in[4];
  const float* b2 = (const float*)d_in[5];
  const float* W3 = (const float*)d_in[6];
  const float* b3 = (const float*)d_in[7];
  float* out = (float*)d_out;

  float* newxyz = (float*)d_ws;  // B*NPOINT*3 floats
  const size_t nxBytes = ((size_t)kB * kNPoint * 3 * sizeof(float) + 255) & ~(size_t)255;
  int* idx = (int*)((char*)d_ws + nxBytes);  // B*NPOINT*NSAMPLE ints (~1 MB)

  fps_kernel<<<kB, 1024, 0, stream>>>(xyz, newxyz);
  ballq_kernel<<<(kB * kNPoint) / 8, 256, 0, stream>>>(xyz, newxyz, idx);
  mlp_kernel<<<1024, 128, 0, stream>>>(xyz, features, newxyz, idx,
                                       W1, b1, W2, b2, W3, b3, out);
}